// MultiHeadSelfAttention_28372553958071
// MI455X (gfx1250) — compile-verified
//
#include <hip/hip_runtime.h>
#include <math.h>

#define B_    128
#define S_    500
#define SP_   512
#define D_    32
#define H_    8
#define HS_   4
#define FCIN_ 16000
#define FC1_  1024
#define FC2_  64

typedef __attribute__((ext_vector_type(2))) float v2f;
typedef __attribute__((ext_vector_type(8))) float v8f;

#if __has_builtin(__builtin_amdgcn_tensor_load_to_lds) && \
    __has_builtin(__builtin_amdgcn_s_wait_tensorcnt)
#define HAVE_TDM 1
typedef __attribute__((ext_vector_type(4))) unsigned int u32x4;
typedef __attribute__((ext_vector_type(8))) int          i32x8;
typedef __attribute__((ext_vector_type(4))) int          i32x4;
#endif

// D = A(16x4) * B(4x16) + C, f32, wave32.
// A layout: lanes 0-15 hold row M=lane, K={0,1}; lanes 16-31 hold row M=lane-16, K={2,3}.
// B layout: lanes 0-15 hold col N=lane, K={0,1}; lanes 16-31 hold col N=lane-16, K={2,3}.
// C/D layout: VGPR r -> M = r + 8*(lane>=16), N = lane&15.
static __device__ __forceinline__ v8f wmma_f32_16x16x4(v2f a, v2f b, v8f c) {
  return __builtin_amdgcn_wmma_f32_16x16x4_f32(false, a, false, b, (short)0, c,
                                               false, false);
}

// ---------------------------------------------------------------------------
// Kernel 1: QKV projections. q/k/v stored as [B,H,S,hs].
// ---------------------------------------------------------------------------
__global__ void qkv_kernel(const float* __restrict__ x,
                           const float* __restrict__ Wq, const float* __restrict__ bq,
                           const float* __restrict__ Wk, const float* __restrict__ bk,
                           const float* __restrict__ Wv, const float* __restrict__ bv,
                           float* __restrict__ qo, float* __restrict__ ko,
                           float* __restrict__ vo) {
  int idx = blockIdx.x * blockDim.x + threadIdx.x;
  if (idx >= B_ * S_ * D_) return;
  int dout = idx & (D_ - 1);
  int bs   = idx >> 5;            // b*S + s
  int s    = bs % S_;
  int b    = bs / S_;
  const float* xr = x + bs * D_;
  const float* wq = Wq + dout * D_;
  const float* wk = Wk + dout * D_;
  const float* wv = Wv + dout * D_;
  float aq = bq[dout], ak = bk[dout], av = bv[dout];
#pragma unroll
  for (int t = 0; t < D_; ++t) {
    float xv = xr[t];
    aq += xv * wq[t];
    ak += xv * wk[t];
    av += xv * wv[t];
  }
  int h = dout >> 2, j = dout & 3;
  int o = ((b * H_ + h) * S_ + s) * HS_ + j;
  qo[o] = aq;
  ko[o] = ak;
  vo[o] = av;
}

// ---------------------------------------------------------------------------
// Kernel 2: attention for one (b, h, 16-query tile) per single-wave block.
// ---------------------------------------------------------------------------
__global__ __launch_bounds__(32) void attn_kernel(const float* __restrict__ qw,
                                                  const float* __restrict__ kw,
                                                  const float* __restrict__ vw,
                                                  float* __restrict__ avo) {
  __shared__ float Kl[SP_ * HS_];       // keys, padded to 512 with zeros
  __shared__ float Vl[SP_ * HS_];
  __shared__ float Sc[16 * SP_];        // score tile [qrow][key]

  int lane = threadIdx.x;
  int qt   = blockIdx.x & 31;           // query tile 0..31
  int bh   = blockIdx.x >> 5;           // b*H + h
  int h    = bh & (H_ - 1);
  int b    = bh >> 3;
  const float* kp = kw + bh * S_ * HS_;
  const float* vp = vw + bh * S_ * HS_;
  const float* qp = qw + bh * S_ * HS_;

  for (int i = lane; i < SP_ * HS_; i += 32) {
    Kl[i] = (i < S_ * HS_) ? kp[i] : 0.f;
    Vl[i] = (i < S_ * HS_) ? vp[i] : 0.f;
  }
  __syncthreads();

  int half = lane >> 4;
  int l15  = lane & 15;

  int srow = qt * 16 + l15;
  if (srow >= S_) srow = S_ - 1;        // clamped reads; garbage rows never stored
  v2f qa;
  qa[0] = qp[srow * HS_ + 2 * half + 0];
  qa[1] = qp[srow * HS_ + 2 * half + 1];

  const float scale = 0.5f;             // 1/sqrt(4)
  for (int kt = 0; kt < SP_ / 16; ++kt) {
    v2f kb;
    kb[0] = Kl[(kt * 16 + l15) * HS_ + 2 * half + 0];
    kb[1] = Kl[(kt * 16 + l15) * HS_ + 2 * half + 1];
    v8f c = {};
    c = wmma_f32_16x16x4(qa, kb, c);
#pragma unroll
    for (int r = 0; r < 8; ++r)
      Sc[(r + 8 * half) * SP_ + kt * 16 + l15] = c[r] * scale;
  }
  __syncthreads();

  // Softmax over the 500 real keys: lane (l15,half) owns row l15, 250-key half.
  {
    float* row = Sc + l15 * SP_ + half * 250;
    float m = -1e30f;
    for (int j = 0; j < 250; ++j) m = fmaxf(m, row[j]);
    m = fmaxf(m, __shfl_xor(m, 16, 32));
    float sum = 0.f;
    for (int j = 0; j < 250; ++j) {
      float e = __expf(row[j] - m);
      row[j]  = e;
      sum += e;
    }
    sum += __shfl_xor(sum, 16, 32);
    float inv = 1.f / sum;
    for (int j = 0; j < 250; ++j) row[j] *= inv;
  }
  __syncthreads();

  // AV: attn(16x500) x V(500x4) as 125 chained 16x16x4 WMMAs (N padded to 16).
  v8f acc = {};
  for (int c4 = 0; c4 < S_ / HS_; ++c4) {
    v2f a, bb;
    a[0] = Sc[l15 * SP_ + 4 * c4 + 2 * half + 0];
    a[1] = Sc[l15 * SP_ + 4 * c4 + 2 * half + 1];
    int key0 = 4 * c4 + 2 * half;
    float v0 = Vl[key0 * HS_ + (l15 & 3)];
    float v1 = Vl[(key0 + 1) * HS_ + (l15 & 3)];
    bool ok = l15 < HS_;                 // select, not branch: EXEC stays full
    bb[0] = ok ? v0 : 0.f;
    bb[1] = ok ? v1 : 0.f;
    acc = wmma_f32_16x16x4(a, bb, acc);
  }

#pragma unroll
  for (int r = 0; r < 8; ++r) {
    int m  = r + 8 * half;
    int sr = qt * 16 + m;
    if (sr < S_ && l15 < HS_)
      avo[(b * S_ + sr) * D_ + h * HS_ + l15] = acc[r];
  }
}

// ---------------------------------------------------------------------------
// Kernel 3: output projection (Wo) -> flat [B, S*D]
// ---------------------------------------------------------------------------
__global__ void proj_kernel(const float* __restrict__ av,
                            const float* __restrict__ Wo,
                            const float* __restrict__ bo,
                            float* __restrict__ flat) {
  int idx = blockIdx.x * blockDim.x + threadIdx.x;
  if (idx >= B_ * S_ * D_) return;
  int dout = idx & (D_ - 1);
  int bs   = idx >> 5;
  const float* ar = av + bs * D_;
  const float* w  = Wo + dout * D_;
  float acc = bo[dout];
#pragma unroll
  for (int t = 0; t < D_; ++t) acc += ar[t] * w[t];
  flat[idx] = acc;  // flat index == (b*S+s)*32 + dout
}

// ---------------------------------------------------------------------------
// Kernel 4: FC1  C[128,1024] = flat[128,16000] @ W1^T + b1.
// 64 blocks x 256 threads; wave w owns a 16x16 output tile at n = nbase+16w.
// W1 tiles (128 rows x 64 cols) are DMA'd into LDS by the Tensor Data Mover
// with hardware row padding (64 DWORDs + 1 pad -> stride 65, conflict-free),
// double-buffered against the WMMA K-loop. Fallback: vector-load staging.
// ---------------------------------------------------------------------------
#define KC_   64
#define NIT_  (FCIN_ / KC_)   /* 250 */

#ifdef HAVE_TDM
__device__ __forceinline__ void tdm_load_w1_tile(const float* W1, int nbase,
                                                 int kb, float* lds_dst) {
  // D# group 0: count=1 | lds_addr | 57-bit global addr | type=2
  unsigned long long ga =
      (unsigned long long)(const void*)(W1 + (size_t)nbase * FCIN_ + kb);
  unsigned lds_addr = (unsigned)(unsigned long long)(void*)lds_dst;  // low 32b = LDS offset
  u32x4 g0;
  g0[0] = 1u;                                   // count=1 valid descriptor
  g0[1] = lds_addr;                             // bits 63:32
  g0[2] = (unsigned)(ga & 0xFFFFFFFFu);         // global addr low
  g0[3] = (unsigned)((ga >> 32) & 0x1FFFFFFu)   // global addr high (57-bit)
          | 0x80000000u;                        // type=2 at bits 127:126
  // D# group 1: data_size=4B, pad 1 DWORD after every 64 DWORDs,
  // tensor 16000 x 1024 (stride 16000), tile 64 x 128.
  i32x8 g1;
  g1[0] = (int)((2u << 16)      // data_size = 4 bytes
                | (1u << 20)    // pad_enable
                | (5u << 22)    // pad_interval = 64 DWORDs
                | (0u << 25));  // pad_amount  = 1 DWORD
  g1[1] = (int)((unsigned)(FCIN_ & 0xFFFF) << 16);  // tensor_dim0 lo16 (barrier addr=0)
  g1[2] = (int)((unsigned)(FC1_ & 0xFFFF) << 16);   // dim0 hi16=0 | tensor_dim1 lo16
  g1[3] = (int)(64u << 16);                         // dim1 hi16=0 | tile_dim0=64
  g1[4] = 128;                                      // tile_dim1=128 | tile_dim2=0
  g1[5] = FCIN_;                                    // tensor_dim0_stride lo32
  g1[6] = 0;                                        // stride hi16 | dim1_stride lo16
  g1[7] = 0;
  i32x4 gz4 = {0, 0, 0, 0};                         // 2D tile: groups 2/3 unused
  i32x8 gz8 = {0, 0, 0, 0, 0, 0, 0, 0};
  // amdgpu-toolchain (clang-23) 6-arg form.
  __builtin_amdgcn_tensor_load_to_lds(g0, g1, gz4, gz4, gz8, 0);
}
#endif

__global__ __launch_bounds__(256) void fc1_kernel(const float* __restrict__ A,
                                                  const float* __restrict__ W1,
                                                  const float* __restrict__ b1,
                                                  float* __restrict__ C) {
  __shared__ float Al[2][16 * 65];   // padded row stride 65 -> conflict-free
  __shared__ float Bl[2][128 * 65];
  int tid  = threadIdx.x;
  int lane = tid & 31, wave = tid >> 5;
  int mtile = blockIdx.x >> 3;     // 0..7  (M = 128)
  int nblk  = blockIdx.x & 7;      // 0..7  (N = 1024, 128 per block)
  int mbase = mtile * 16;
  int nbase = nblk * 128;
  int half = lane >> 4, l15 = lane & 15;

  v8f acc = {};

#ifdef HAVE_TDM
  if (wave == 0) tdm_load_w1_tile(W1, nbase, 0, &Bl[0][0]);
#endif

  for (int it = 0; it < NIT_; ++it) {
    int kb  = it * KC_;
    int buf = it & 1;

    // Stage the A tile (16 x 64) with regular vector loads.
    for (int i = tid; i < 16 * KC_; i += 256) {
      int r = i >> 6, cc = i & 63;
      Al[buf][r * 65 + cc] = A[(mbase + r) * FCIN_ + kb + cc];
    }

#ifdef HAVE_TDM
    if (wave == 0) {
      if (it + 1 < NIT_) {
        tdm_load_w1_tile(W1, nbase, kb + KC_, &Bl[buf ^ 1][0]);
        __builtin_amdgcn_s_wait_tensorcnt(1);  // tile `it` landed; next in flight
      } else {
        __builtin_amdgcn_s_wait_tensorcnt(0);
      }
    }
#else
    for (int i = tid; i < 128 * KC_; i += 256) {
      int r = i >> 6, cc = i & 63;
      Bl[buf][r * 65 + cc] = W1[(nbase + r) * FCIN_ + kb + cc];
    }
    if (kb + KC_ < FCIN_) {        // prefetch next W1 tile (global_prefetch_b8)
      int r = tid >> 1, cc = (tid & 1) * 32;
      __builtin_prefetch(&W1[(nbase + r) * FCIN_ + kb + KC_ + cc], 0, 0);
    }
#endif
    __syncthreads();

    const float* Aw = &Al[buf][l15 * 65];
    const float* Bw = &Bl[buf][(wave * 16 + l15) * 65];
#pragma unroll
    for (int kc = 0; kc < KC_ / 4; ++kc) {
      int k4 = kc * 4 + 2 * half;
      v2f a, bb;
      a[0]  = Aw[k4];
      a[1]  = Aw[k4 + 1];
      bb[0] = Bw[k4];
      bb[1] = Bw[k4 + 1];
      acc = wmma_f32_16x16x4(a, bb, acc);
    }
    __syncthreads();
  }

  int n = nbase + wave * 16 + l15;
  float bias = b1[n];
#pragma unroll
  for (int r = 0; r < 8; ++r) {
    int m = mbase + r + 8 * half;
    C[m * FC1_ + n] = acc[r] + bias;
  }
}

// ---------------------------------------------------------------------------
// Kernel 5: FC2 + FC3 fused per batch row. Writes both tuple outputs.
// ---------------------------------------------------------------------------
__global__ __launch_bounds__(64) void fc23_kernel(const float* __restrict__ f1,
                                                  const float* __restrict__ W2,
                                                  const float* __restrict__ b2,
                                                  const float* __restrict__ W3,
                                                  const float* __restrict__ b3,
                                                  float* __restrict__ out1,
                                                  float* __restrict__ out2) {
  __shared__ float s1[64];
  int i = blockIdx.x, t = threadIdx.x;
  const float* row = f1 + i * FC1_;
  const float* w   = W2 + t * FC1_;
  float acc = b2[t];
  for (int k = 0; k < FC1_; ++k) acc += row[k] * w[k];
  out1[i * 64 + t] = acc;
  s1[t] = acc;
  __syncthreads();
  if (t < 2) {
    float a3 = b3[t];
    const float* w3 = W3 + t * 64;
#pragma unroll
    for (int k = 0; k < 64; ++k) a3 += s1[k] * w3[k];
    out2[i * 2 + t] = a3;
  }
}

// ---------------------------------------------------------------------------
extern "C" void kernel_launch(void* const* d_in, const int* in_sizes, int n_in,
                              void* d_out, int out_size, void* d_ws, size_t ws_size,
                              hipStream_t stream) {
  const float* x  = (const float*)d_in[0];
  const float* Wq = (const float*)d_in[1];
  const float* bq = (const float*)d_in[2];
  const float* Wk = (const float*)d_in[3];
  const float* bk = (const float*)d_in[4];
  const float* Wv = (const float*)d_in[5];
  const float* bv = (const float*)d_in[6];
  const float* Wo = (const float*)d_in[7];
  const float* bo = (const float*)d_in[8];
  const float* W1 = (const float*)d_in[9];
  const float* b1 = (const float*)d_in[10];
  const float* W2 = (const float*)d_in[11];
  const float* b2 = (const float*)d_in[12];
  const float* W3 = (const float*)d_in[13];
  const float* b3 = (const float*)d_in[14];

  float* ws   = (float*)d_ws;
  float* q    = ws;                       // [B,H,S,hs] 2,048,000 f
  float* k    = ws + 2048000;
  float* v    = ws + 4096000;
  float* av   = ws + 6144000;             // [B,S,D]
  float* flat = ws + 8192000;             // [B, S*D]
  float* f1   = ws + 10240000;            // [B, 1024]

  float* out1 = (float*)d_out;            // [128,64]
  float* out2 = out1 + B_ * FC2_;         // [128,2]

  int N = B_ * S_ * D_;
  qkv_kernel<<<(N + 255) / 256, 256, 0, stream>>>(x, Wq, bq, Wk, bk, Wv, bv, q, k, v);
  attn_kernel<<<B_ * H_ * 32, 32, 0, stream>>>(q, k, v, av);
  proj_kernel<<<(N + 255) / 256, 256, 0, stream>>>(av, Wo, bo, flat);
  fc1_kernel<<<64, 256, 0, stream>>>(flat, W1, b1, f1);
  fc23_kernel<<<B_, 64, 0, stream>>>(f1, W2, b2, W3, b3, out1, out2);
}